// SparseAttention_3118146257661
// MI455X (gfx1250) — compile-verified
//
#include <hip/hip_runtime.h>
#include <hip/hip_bf16.h>

typedef __attribute__((ext_vector_type(2))) float v2f;
typedef __attribute__((ext_vector_type(8))) float v8f;

#define NTOK 512
#define DIN  128
#define TOPK 12

__launch_bounds__(256)
__global__ void spattn_kernel(const float* __restrict__ x,
                              const float* __restrict__ wkg,
                              const float* __restrict__ wqg,
                              float* __restrict__ out) {
  __shared__ float s_wk[DIN * 4];
  __shared__ float s_wq[DIN * 4];
  __shared__ alignas(16) float s_k[NTOK * 4];
  __shared__ alignas(16) float s_q[NTOK * 4];
  __shared__ alignas(16) float s_Ap[8 * NTOK];   // per-wave partial column sums
  __shared__ alignas(16) float s_A[NTOK];
  __shared__ int   s_idx[TOPK];

  const int b     = blockIdx.x;
  const int tid   = threadIdx.x;
  const int lane  = tid & 31;
  const int wave  = tid >> 5;
  const int laneN = lane & 15;
  const int hi    = lane >> 4;              // 0: rows v, K=0..1 ; 1: rows v+8, K=2..3
  const float scale = 0.08838834764831845f; // 1/sqrt(128)

  const float* xb = x + (size_t)b * (NTOK * DIN);

  for (int i = tid; i < DIN * 4; i += 256) { s_wk[i] = wkg[i]; s_wq[i] = wqg[i]; }
  for (int i = tid; i < 8 * NTOK; i += 256) s_Ap[i] = 0.0f;
  __syncthreads();

  // ---- Phase 1: k = x@wk, q = x@wq  ([512,4] each, into LDS) ----
  for (int n = tid; n < NTOK; n += 256) {
    const float* xr = xb + n * DIN;
    float kk[4] = {0.f, 0.f, 0.f, 0.f};
    float qq[4] = {0.f, 0.f, 0.f, 0.f};
    for (int d = 0; d < DIN; d += 4) {
      float4 xv = *(const float4*)(xr + d);
#pragma unroll
      for (int e = 0; e < 4; ++e) {
        kk[e] += xv.x * s_wk[(d + 0) * 4 + e];
        kk[e] += xv.y * s_wk[(d + 1) * 4 + e];
        kk[e] += xv.z * s_wk[(d + 2) * 4 + e];
        kk[e] += xv.w * s_wk[(d + 3) * 4 + e];
        qq[e] += xv.x * s_wq[(d + 0) * 4 + e];
        qq[e] += xv.y * s_wq[(d + 1) * 4 + e];
        qq[e] += xv.z * s_wq[(d + 2) * 4 + e];
        qq[e] += xv.w * s_wq[(d + 3) * 4 + e];
      }
    }
#pragma unroll
    for (int e = 0; e < 4; ++e) { s_k[n * 4 + e] = kk[e]; s_q[n * 4 + e] = qq[e]; }
  }
  __syncthreads();

  // ---- Phase 2: h = k q^T via V_WMMA_F32_16X16X4_F32, fused row-softmax ----
  // Each wave handles 16-row blocks rb = wave, wave+8, ...
  for (int rb = wave; rb < 32; rb += 8) {
    const int n0 = rb * 16;
    // A operand, 16x4 f32: lane<16 -> row n0+laneN K=0,1 ; lane>=16 -> K=2,3
    v2f a = *(const v2f*)(&s_k[(n0 + laneN) * 4 + hi * 2]);

    // Pass 1: row maxima
    float rm[8];
#pragma unroll
    for (int v = 0; v < 8; ++v) rm[v] = -__builtin_inff();
    for (int mt = 0; mt < 32; ++mt) {
      v2f bb = *(const v2f*)(&s_q[(mt * 16 + laneN) * 4 + hi * 2]);
      v8f c = {};
      v8f d = __builtin_amdgcn_wmma_f32_16x16x4_f32(false, a, false, bb, (short)0, c, false, false);
#pragma unroll
      for (int v = 0; v < 8; ++v) rm[v] = fmaxf(rm[v], d[v]);
    }
#pragma unroll
    for (int off = 1; off <= 8; off <<= 1) {
#pragma unroll
      for (int v = 0; v < 8; ++v) rm[v] = fmaxf(rm[v], __shfl_xor(rm[v], off, 32));
    }

    // Pass 2: row sums of exp(scale*(h - rowmax))
    float rs[8];
#pragma unroll
    for (int v = 0; v < 8; ++v) rs[v] = 0.f;
    for (int mt = 0; mt < 32; ++mt) {
      v2f bb = *(const v2f*)(&s_q[(mt * 16 + laneN) * 4 + hi * 2]);
      v8f c = {};
      v8f d = __builtin_amdgcn_wmma_f32_16x16x4_f32(false, a, false, bb, (short)0, c, false, false);
#pragma unroll
      for (int v = 0; v < 8; ++v) rs[v] += __expf(scale * (d[v] - rm[v]));
    }
#pragma unroll
    for (int off = 1; off <= 8; off <<= 1) {
#pragma unroll
      for (int v = 0; v < 8; ++v) rs[v] += __shfl_xor(rs[v], off, 32);
    }
    float rinv[8];
#pragma unroll
    for (int v = 0; v < 8; ++v) rinv[v] = 1.0f / rs[v];

    // Pass 3: accumulate column partials A[m] += sum_n softmax[n][m]
    for (int mt = 0; mt < 32; ++mt) {
      v2f bb = *(const v2f*)(&s_q[(mt * 16 + laneN) * 4 + hi * 2]);
      v8f c = {};
      v8f d = __builtin_amdgcn_wmma_f32_16x16x4_f32(false, a, false, bb, (short)0, c, false, false);
      float ssum = 0.f;
#pragma unroll
      for (int v = 0; v < 8; ++v) ssum += __expf(scale * (d[v] - rm[v])) * rinv[v];
      ssum += __shfl_xor(ssum, 16, 32);        // combine rows 0-7 and 8-15 halves
      if (hi == 0)
        s_Ap[wave * NTOK + mt * 16 + laneN] += ssum;  // exclusive per (wave,col): deterministic
    }
  }
  __syncthreads();

  // ---- Reduce per-wave partials into A (fixed order => deterministic) ----
  for (int m = tid; m < NTOK; m += 256) {
    float acc = 0.f;
#pragma unroll
    for (int w = 0; w < 8; ++w) acc += s_Ap[w * NTOK + m];
    s_A[m] = acc;
  }
  __syncthreads();

  // ---- Top-12 by rank counting (matches lax.top_k: descending, index tie-break) ----
  // Each thread owns m0 = tid and m1 = tid + 256; single vectorized sweep over j.
  {
    const int m0 = tid;
    const int m1 = tid + 256;
    const float a0 = s_A[m0];
    const float a1 = s_A[m1];
    int r0 = 0, r1 = 0;
    for (int j = 0; j < NTOK; j += 4) {
      float4 aj = *(const float4*)(&s_A[j]);
      r0 += (aj.x > a0) || ((aj.x == a0) && ((j + 0) < m0));
      r0 += (aj.y > a0) || ((aj.y == a0) && ((j + 1) < m0));
      r0 += (aj.z > a0) || ((aj.z == a0) && ((j + 2) < m0));
      r0 += (aj.w > a0) || ((aj.w == a0) && ((j + 3) < m0));
      r1 += (aj.x > a1) || ((aj.x == a1) && ((j + 0) < m1));
      r1 += (aj.y > a1) || ((aj.y == a1) && ((j + 1) < m1));
      r1 += (aj.z > a1) || ((aj.z == a1) && ((j + 2) < m1));
      r1 += (aj.w > a1) || ((aj.w == a1) && ((j + 3) < m1));
    }
    if (r0 < TOPK) s_idx[r0] = m0;
    if (r1 < TOPK) s_idx[r1] = m1;
  }
  __syncthreads();

  // ---- Gather selected rows: out[b, j, :] = x[b, idx[j], :] ----
  float* ob = out + (size_t)b * (TOPK * DIN);
  for (int i = tid; i < TOPK * (DIN / 4); i += 256) {
    int j  = i / (DIN / 4);
    int d4 = i % (DIN / 4);
    int src = s_idx[j];
    *(float4*)(ob + j * DIN + d4 * 4) = *(const float4*)(xb + src * DIN + d4 * 4);
  }
}

extern "C" void kernel_launch(void* const* d_in, const int* in_sizes, int n_in,
                              void* d_out, int out_size, void* d_ws, size_t ws_size,
                              hipStream_t stream) {
  const float* x  = (const float*)d_in[0];
  const float* wk = (const float*)d_in[1];
  const float* wq = (const float*)d_in[2];
  float* out = (float*)d_out;
  (void)in_sizes; (void)n_in; (void)out_size; (void)d_ws; (void)ws_size;
  spattn_kernel<<<dim3(512), dim3(256), 0, stream>>>(x, wk, wq, out);
}